// ReLU15Loss_82695300317293
// MI455X (gfx1250) — compile-verified
//
#include <hip/hip_runtime.h>
#include <hip/hip_bf16.h>

// ---------------------------------------------------------------------------
// ReLU15 / Tsallis-1.5 loss, MI455X (gfx1250).
// Memory-bound streaming reduction: 1.05 GB read -> ~45us floor @ 23.3 TB/s.
// CDNA5 async-tensor data path: 4-deep global_load_async_to_lds_b128 ring
// (2 KB in flight per wave) + s_wait_asynccnt, consumed via ds_load_b128.
// Element math: t = relu(0.5x - tau); p = t^2; p^1.5 = t^3 exactly, so each
// element contributes t^2 * ((1-t)*4/3 + x - 2*tau)  -> ~5 packed FMAs.
// ---------------------------------------------------------------------------

#define NCLS      32000
#define NF4       8000        // float4 per row
#define BS        256         // threads per block (8 wave32)
#define MAIN_IT   31          // 31*256 = 7936 float4 handled in async pipeline
#define TAIL_F4   (NF4 - MAIN_IT * BS)   // 64 float4 tail
#define DEPTH     4           // async ring depth

__device__ __forceinline__ float relu15_contrib(float x) {
    float t  = fmaf(0.5f, x, -0.05f);
    t        = fmaxf(t, 0.0f);
    float t2 = t * t;
    float g  = fmaf(-1.33333333333333f, t, 1.33333333333333f) + (x - 0.1f);
    return t2 * g;
}

__global__ __launch_bounds__(BS)
void relu15_row_kernel(const float* __restrict__ X,
                       const int*   __restrict__ target,
                       float*       __restrict__ row_loss) {
    __shared__ float4 abuf[DEPTH][BS];   // 16 KB async ring (per-lane slots)
    __shared__ float  wred[BS / 32];

    const int tid = threadIdx.x;
    const int row = blockIdx.x;
    const float* rowp = X + (size_t)row * NCLS;

    // Raw LDS byte offsets of this lane's ring slots (low 32 bits of the
    // generic LDS address == workgroup-relative LDS offset on gfx1250).
    unsigned lds[DEPTH];
#pragma unroll
    for (int b = 0; b < DEPTH; ++b)
        lds[b] = (unsigned)(size_t)&abuf[b][tid];

    // Prime the pipeline: async-load iterations 0..2 (ASYNCcnt=3).
#pragma unroll
    for (int it = 0; it < DEPTH - 1; ++it) {
        unsigned goff = (unsigned)(tid + it * BS) * 16u;
        asm volatile("global_load_async_to_lds_b128 %0, %1, %2"
                     :: "v"(lds[it]), "v"(goff), "s"(rowp) : "memory");
    }

    float acc = 0.0f;
#pragma unroll 1
    for (int it = 0; it < MAIN_IT; ++it) {
        if (it + DEPTH - 1 < MAIN_IT) {
            // Prefetch tile it+3 into the slot last read at iteration it-1.
            // s_wait_dscnt 0 closes the WAR hazard (async LDS writes are
            // unordered vs. ds_load); then ASYNCcnt<=3 guarantees the oldest
            // outstanding load -- the one for buffer it&3 -- has landed.
            const unsigned nxt  = lds[(it + DEPTH - 1) % DEPTH];
            const unsigned goff = (unsigned)(tid + (it + DEPTH - 1) * BS) * 16u;
            asm volatile("s_wait_dscnt 0x0\n\t"
                         "global_load_async_to_lds_b128 %0, %1, %2"
                         :: "v"(nxt), "v"(goff), "s"(rowp) : "memory");
            asm volatile("s_wait_asynccnt 0x3" ::: "memory");
        } else {
            // Last DEPTH-1 iterations: everything already issued; drain.
            asm volatile("s_wait_asynccnt 0x0" ::: "memory");
        }
        const float4 v = abuf[it % DEPTH][tid];   // ds_load_b128
        acc += relu15_contrib(v.x);
        acc += relu15_contrib(v.y);
        acc += relu15_contrib(v.z);
        acc += relu15_contrib(v.w);
    }

    // Tail: 64 remaining float4 (threads 0..63), plain B128 loads.
    if (tid < TAIL_F4) {
        const float4* p4 = (const float4*)rowp;
        const float4 v = p4[MAIN_IT * BS + tid];
        acc += relu15_contrib(v.x);
        acc += relu15_contrib(v.y);
        acc += relu15_contrib(v.z);
        acc += relu15_contrib(v.w);
    }

    // Wave32 tree reduce (fixed order -> deterministic).
#pragma unroll
    for (int off = 16; off > 0; off >>= 1)
        acc += __shfl_down(acc, off, 32);
    if ((tid & 31) == 0) wred[tid >> 5] = acc;
    __syncthreads();

    if (tid == 0) {
        float s = 0.0f;
#pragma unroll
        for (int w = 0; w < BS / 32; ++w) s += wred[w];
        // - (x_target - 2*tau) from the scatter of -1 at the target column.
        const int t = target[row];
        s -= (rowp[t] - 0.1f);
        row_loss[row] = s;
    }
}

__global__ __launch_bounds__(BS)
void relu15_final_reduce(const float* __restrict__ row_loss,
                         float*       __restrict__ out,
                         int n_rows) {
    __shared__ double red[BS];
    double s = 0.0;
    for (int r = threadIdx.x; r < n_rows; r += BS)
        s += (double)row_loss[r];
    red[threadIdx.x] = s;
    __syncthreads();
#pragma unroll
    for (int off = BS / 2; off > 0; off >>= 1) {
        if (threadIdx.x < off) red[threadIdx.x] += red[threadIdx.x + off];
        __syncthreads();
    }
    if (threadIdx.x == 0)
        out[0] = (float)(red[0] / (double)n_rows);
}

extern "C" void kernel_launch(void* const* d_in, const int* in_sizes, int n_in,
                              void* d_out, int out_size, void* d_ws, size_t ws_size,
                              hipStream_t stream) {
    const float* X      = (const float*)d_in[0];
    const int*   target = (const int*)d_in[1];
    float*       out    = (float*)d_out;
    float*       rloss  = (float*)d_ws;    // n_rows floats of scratch

    const int n_rows = in_sizes[1];        // 8192

    relu15_row_kernel<<<n_rows, BS, 0, stream>>>(X, target, rloss);
    relu15_final_reduce<<<1, BS, 0, stream>>>(rloss, out, n_rows);
}